// TinyGPT_9045201125979
// MI455X (gfx1250) — compile-verified
//
#include <hip/hip_runtime.h>
#include <hip/hip_bf16.h>
#include <math.h>
#include <stdint.h>

// ---------------------------------------------------------------------------
// TinyGPT forward on gfx1250 (CDNA5), f32 WMMA (V_WMMA_F32_16X16X4_F32) with
// ASYNCcnt-based double-buffered global->LDS staging in the GEMMs.
// B=4 T=2048 V=32000 D=1024.
// ---------------------------------------------------------------------------

typedef float v2f __attribute__((ext_vector_type(2)));
typedef float v8f __attribute__((ext_vector_type(8)));
typedef int   v4i __attribute__((ext_vector_type(4)));

#define WMMA_F32(a, b, c) \
  __builtin_amdgcn_wmma_f32_16x16x4_f32(false, (a), false, (b), (short)0, (c), false, false)

static constexpr int B_ = 4, T_ = 2048, V_ = 32000, D_ = 1024;
static constexpr int M_ = B_ * T_;  // 8192 rows

// ---- async global->LDS support (guarded so absence never breaks the build) --
#if defined(__AMDGCN__) && \
    __has_builtin(__builtin_amdgcn_global_load_async_to_lds_b128) && \
    __has_builtin(__builtin_amdgcn_s_wait_asynccnt)
#define HAVE_ASYNC_LDS 1
#else
#define HAVE_ASYNC_LDS 0
#endif

#if HAVE_ASYNC_LDS
typedef __attribute__((address_space(1))) v4i g1_v4i;  // global int4
typedef __attribute__((address_space(3))) v4i l3_v4i;  // LDS int4
#define ASYNC_WAIT() __builtin_amdgcn_s_wait_asynccnt(0)
__device__ __forceinline__ void async_copy16(void* lds_dst, const void* gsrc) {
  __builtin_amdgcn_global_load_async_to_lds_b128(
      (g1_v4i*)(uintptr_t)gsrc, (l3_v4i*)(uint32_t)(uintptr_t)lds_dst, 0, 0);
}
#else
#define ASYNC_WAIT()
__device__ __forceinline__ void async_copy16(void* lds_dst, const void* gsrc) {
  *(float4*)lds_dst = *(const float4*)gsrc;
}
#endif

// ---------------------------------------------------------------------------
// Kernel 1: x[b,t,:] = tok_emb[index[b,t],:] + pos_emb[t,:]
// ---------------------------------------------------------------------------
__global__ __launch_bounds__(256) void tg_embed(
    const int* __restrict__ index, const float* __restrict__ tok_emb,
    const float* __restrict__ pos_emb, float* __restrict__ x) {
  const size_t base = ((size_t)blockIdx.x * 256 + threadIdx.x) * 4;  // float4 granularity
  if (base >= (size_t)M_ * D_) return;
  const int bt = (int)(base / D_);
  const int d  = (int)(base % D_);
  const int t  = bt % T_;
  const int tok = index[bt];
  const float4 te = *(const float4*)(tok_emb + (size_t)tok * D_ + d);
  const float4 pe = *(const float4*)(pos_emb + (size_t)t * D_ + d);
  float4 o;
  o.x = te.x + pe.x; o.y = te.y + pe.y; o.z = te.z + pe.z; o.w = te.w + pe.w;
  *(float4*)(x + base) = o;
}

// ---------------------------------------------------------------------------
// Kernel 2: generic C[M,N] = A[M,K] @ B[K,N] (+bias), row-major f32.
// Workgroup tile 64x128, 8 waves, each wave a 32x32 quad of 16x16 WMMA tiles.
// Double-buffered LDS with async global->LDS copies (ASYNCcnt).
// ---------------------------------------------------------------------------
#define GTM 64
#define GTN 128
#define GKC 32
#define GPAD 4  // keeps rows 16B-aligned (stride 36 floats = 144B) + bank-spread

__global__ __launch_bounds__(256) void tg_gemm_f32(
    const float* __restrict__ A, const float* __restrict__ B,
    const float* __restrict__ bias, float* __restrict__ C,
    int M, int N, int K) {
  __shared__ float As[2][GTM][GKC + GPAD];  // 2 x 9.2KB
  __shared__ float Bs[2][GKC][GTN];         // 2 x 16KB

  const int tid  = threadIdx.x;
  const int lane = tid & 31;
  const int wave = tid >> 5;
  const int lr   = lane & 15;
  const int kh   = (lane < 16) ? 0 : 2;   // f32 WMMA A/B half-wave K offset
  const int tm0  = (wave >> 2) * 32;      // 0 or 32
  const int tn0  = (wave & 3) * 32;       // 0,32,64,96
  const int gm   = blockIdx.y * GTM;
  const int gn   = blockIdx.x * GTN;

  auto copy_tiles = [&](int k0, int buf) {
    // A tile: 64x32 floats = 512 16B chunks, 2 per thread
#pragma unroll
    for (int i = tid; i < GTM * GKC / 4; i += 256) {
      const int r = i >> 3, c4 = (i & 7) * 4;
      async_copy16(&As[buf][r][c4], A + (size_t)(gm + r) * K + k0 + c4);
    }
    // B tile: 32x128 floats = 1024 16B chunks, 4 per thread
#pragma unroll
    for (int i = tid; i < GKC * GTN / 4; i += 256) {
      const int r = i >> 5, c4 = (i & 31) * 4;
      async_copy16(&Bs[buf][r][c4], B + (size_t)(k0 + r) * N + gn + c4);
    }
  };

  v8f acc[2][2];
#pragma unroll
  for (int im = 0; im < 2; ++im)
#pragma unroll
    for (int in = 0; in < 2; ++in)
      acc[im][in] = (v8f){0.f, 0.f, 0.f, 0.f, 0.f, 0.f, 0.f, 0.f};

  copy_tiles(0, 0);
  ASYNC_WAIT();
  __syncthreads();

  int cur = 0;
  for (int k0 = 0; k0 < K; k0 += GKC) {
    // stage next chunk while computing this one
    if (k0 + GKC < K) copy_tiles(k0 + GKC, cur ^ 1);

    const float (*as)[GKC + GPAD] = As[cur];
    const float (*bs)[GTN]        = Bs[cur];
#pragma unroll
    for (int kk = 0; kk < GKC; kk += 4) {
#pragma unroll
      for (int im = 0; im < 2; ++im) {
        v2f a;
        a.x = as[tm0 + im * 16 + lr][kk + kh];
        a.y = as[tm0 + im * 16 + lr][kk + kh + 1];
#pragma unroll
        for (int in = 0; in < 2; ++in) {
          v2f bb;
          bb.x = bs[kk + kh][tn0 + in * 16 + lr];
          bb.y = bs[kk + kh + 1][tn0 + in * 16 + lr];
          acc[im][in] = WMMA_F32(a, bb, acc[im][in]);
        }
      }
    }
    ASYNC_WAIT();     // next chunk's copies have landed
    __syncthreads();  // everyone done reading cur before it becomes the target
    cur ^= 1;
  }

#pragma unroll
  for (int im = 0; im < 2; ++im)
#pragma unroll
    for (int in = 0; in < 2; ++in)
#pragma unroll
      for (int r = 0; r < 8; ++r) {
        const int row = gm + tm0 + im * 16 + ((lane < 16) ? r : r + 8);
        const int col = gn + tn0 + in * 16 + lr;
        float o = acc[im][in][r];
        if (bias) o += bias[col];
        C[(size_t)row * N + col] = o;
      }
}

// ---------------------------------------------------------------------------
// Kernel 3: causal attention, flash-style online softmax.
// One workgroup per (16-query tile, batch). 8 waves split D into 128-wide
// slices: each wave computes partial QK^T over its slice (reduced in LDS),
// then accumulates P @ V over its slice. q == k (source bug).
// ---------------------------------------------------------------------------
__global__ __launch_bounds__(256) void tg_attn(
    const float* __restrict__ kbuf, const float* __restrict__ vbuf,
    float* __restrict__ obuf) {
  __shared__ float Sp[8][256];   // per-wave partial scores
  __shared__ float Stile[256];   // reduced masked scores
  __shared__ float Ptile[256];   // exp probabilities
  __shared__ float mrow[16], lrow[16], crow[16];

  const int qi   = blockIdx.x;   // query tile 0..127
  const int b    = blockIdx.y;   // batch 0..3
  const int tid  = threadIdx.x;
  const int lane = tid & 31;
  const int wave = tid >> 5;
  const int lr   = lane & 15;
  const int kh   = (lane < 16) ? 0 : 2;
  const int dsl  = wave * 128;   // this wave's D-slice
  const float scale = 0.03125f;  // 1024^-0.5
  const int rowbase = b * T_ + qi * 16;

  // preload this wave's q fragments (32 k-steps over its 128-wide slice)
  v2f qa[32];
#pragma unroll
  for (int s = 0; s < 32; ++s) {
    const float* qp = kbuf + (size_t)(rowbase + lr) * D_ + dsl + s * 4 + kh;
    qa[s].x = qp[0];
    qa[s].y = qp[1];
  }

  if (tid < 16) { mrow[tid] = -INFINITY; lrow[tid] = 0.f; }

  v8f acc[8];
#pragma unroll
  for (int i = 0; i < 8; ++i) acc[i] = (v8f){0.f, 0.f, 0.f, 0.f, 0.f, 0.f, 0.f, 0.f};

  __syncthreads();

  for (int j = 0; j <= qi; ++j) {
    const int kb = b * T_ + j * 16;

    // partial S = q_slice @ k_slice^T   (B frag: B[K=d][N=key] = k[key][d])
    v8f sc = (v8f){0.f, 0.f, 0.f, 0.f, 0.f, 0.f, 0.f, 0.f};
#pragma unroll
    for (int s = 0; s < 32; ++s) {
      const float* kp = kbuf + (size_t)(kb + lr) * D_ + dsl + s * 4 + kh;
      v2f bf; bf.x = kp[0]; bf.y = kp[1];
      sc = WMMA_F32(qa[s], bf, sc);
    }
#pragma unroll
    for (int r = 0; r < 8; ++r) {
      const int row = (lane < 16) ? r : r + 8;
      Sp[wave][row * 16 + lr] = sc[r];
    }
    __syncthreads();

    // reduce 8 partials, scale, causal mask
    {
      float s = 0.f;
#pragma unroll
      for (int w = 0; w < 8; ++w) s += Sp[w][tid];
      s *= scale;
      const int row = tid >> 4, col = tid & 15;
      if (j * 16 + col > qi * 16 + row) s = -INFINITY;
      Stile[tid] = s;
    }
    __syncthreads();

    // online softmax row stats
    if (tid < 16) {
      float mx = Stile[tid * 16];
#pragma unroll
      for (int c = 1; c < 16; ++c) mx = fmaxf(mx, Stile[tid * 16 + c]);
      const float mnew = fmaxf(mrow[tid], mx);
      const float corr = __expf(mrow[tid] - mnew);  // exp(-inf)=0 on first tile
      crow[tid] = corr;
      lrow[tid] *= corr;
      mrow[tid] = mnew;
    }
    __syncthreads();

    Ptile[tid] = __expf(Stile[tid] - mrow[tid >> 4]);
    __syncthreads();

    if (tid < 16) {
      float s = 0.f;
#pragma unroll
      for (int c = 0; c < 16; ++c) s += Ptile[tid * 16 + c];
      lrow[tid] += s;
    }

    // rescale accumulator, then acc += P @ V_slice
    float cr[8];
#pragma unroll
    for (int r = 0; r < 8; ++r) cr[r] = crow[(lane < 16) ? r : r + 8];
#pragma unroll
    for (int in = 0; in < 8; ++in)
#pragma unroll
      for (int r = 0; r < 8; ++r) acc[in][r] *= cr[r];

#pragma unroll
    for (int kk = 0; kk < 16; kk += 4) {
      v2f a;
      a.x = Ptile[lr * 16 + kk + kh];
      a.y = Ptile[lr * 16 + kk + kh + 1];
      const float* vp = vbuf + (size_t)(kb + kk + kh) * D_ + dsl;
#pragma unroll
      for (int in = 0; in < 8; ++in) {
        v2f bf;
        bf.x = vp[in * 16 + lr];
        bf.y = vp[D_ + in * 16 + lr];
        acc[in] = WMMA_F32(a, bf, acc[in]);
      }
    }
    __syncthreads();  // protect Sp/Stile/Ptile for next key tile
  }

  __syncthreads();
  float inv[8];
#pragma unroll
  for (int r = 0; r < 8; ++r) inv[r] = 1.f / lrow[(lane < 16) ? r : r + 8];
#pragma unroll
  for (int in = 0; in < 8; ++in)
#pragma unroll
    for (int r = 0; r < 8; ++r) {
      const int row = rowbase + ((lane < 16) ? r : r + 8);
      obuf[(size_t)row * D_ + dsl + in * 16 + lr] = acc[in][r] * inv[r];
    }
}

// ---------------------------------------------------------------------------
// Kernel 4: cross-entropy. One block per row: max pass, sum-exp pass,
// atomic accumulation of -logp(target)/8192 into *loss.
// ---------------------------------------------------------------------------
__global__ __launch_bounds__(256) void tg_ce(
    const float* __restrict__ logits, const int* __restrict__ targets,
    float* __restrict__ loss) {
  __shared__ float red[256];
  const int row = blockIdx.x;
  const float* lp = logits + (size_t)row * V_;
  const int tid = threadIdx.x;

  float mx = -INFINITY;
  for (int c = tid; c < V_; c += 256) mx = fmaxf(mx, lp[c]);
  red[tid] = mx;
  __syncthreads();
  for (int s = 128; s > 0; s >>= 1) {
    if (tid < s) red[tid] = fmaxf(red[tid], red[tid + s]);
    __syncthreads();
  }
  mx = red[0];
  __syncthreads();

  float sum = 0.f;
  for (int c = tid; c < V_; c += 256) sum += __expf(lp[c] - mx);
  red[tid] = sum;
  __syncthreads();
  for (int s = 128; s > 0; s >>= 1) {
    if (tid < s) red[tid] += red[tid + s];
    __syncthreads();
  }

  if (tid == 0) {
    const float lse = mx + __logf(red[0]);
    const float lpt = lp[targets[row]] - lse;
    atomicAdd(loss, -lpt / (float)(B_ * T_));
  }
}

// ---------------------------------------------------------------------------
// Launcher
// ---------------------------------------------------------------------------
extern "C" void kernel_launch(void* const* d_in, const int* in_sizes, int n_in,
                              void* d_out, int out_size, void* d_ws, size_t ws_size,
                              hipStream_t stream) {
  const int*   index   = (const int*)d_in[0];
  const int*   targets = (const int*)d_in[1];
  const float* tok_emb = (const float*)d_in[2];
  const float* pos_emb = (const float*)d_in[3];
  const float* Wk      = (const float*)d_in[4];
  // d_in[5] = Wq : unused (reference bug: q = x @ Wk)
  const float* Wv      = (const float*)d_in[6];
  const float* lm_W    = (const float*)d_in[7];
  const float* lm_b    = (const float*)d_in[8];

  float* logits = (float*)d_out;
  float* loss   = logits + (size_t)M_ * V_;

  float* xbuf = (float*)d_ws;                 // 33.5 MB: x, later reused for attn out
  float* kbuf = xbuf + (size_t)M_ * D_;       // 33.5 MB: k (== q)
  float* vbuf = kbuf + (size_t)M_ * D_;       // 33.5 MB: v

  // 1) embeddings
  tg_embed<<<(M_ * D_ / 4 + 255) / 256, 256, 0, stream>>>(index, tok_emb, pos_emb, xbuf);

  // 2) k = x @ Wk ; v = x @ Wv
  dim3 gkv(D_ / GTN, M_ / GTM);
  tg_gemm_f32<<<gkv, 256, 0, stream>>>(xbuf, Wk, nullptr, kbuf, M_, D_, D_);
  tg_gemm_f32<<<gkv, 256, 0, stream>>>(xbuf, Wv, nullptr, vbuf, M_, D_, D_);

  // 3) attention (writes over xbuf)
  tg_attn<<<dim3(T_ / 16, B_), 256, 0, stream>>>(kbuf, vbuf, xbuf);

  // 4) logits = out @ lm_W + lm_b  (straight into d_out)
  dim3 glm(V_ / GTN, M_ / GTM);
  tg_gemm_f32<<<glm, 256, 0, stream>>>(xbuf, lm_W, lm_b, logits, M_, V_, D_);

  // 5) cross-entropy
  (void)hipMemsetAsync(loss, 0, sizeof(float), stream);
  tg_ce<<<M_, 256, 0, stream>>>(logits, targets, loss);
}